// GINModel_81114752352699
// MI455X (gfx1250) — compile-verified
//
#include <hip/hip_runtime.h>

#define NODES  50000
#define EDGES  800000
#define DIM    128
#define LAYERS 4
#define GRAPHS 512

typedef __attribute__((ext_vector_type(2))) float v2f;
typedef __attribute__((ext_vector_type(8))) float v8f;

__global__ void zero_kernel(float* __restrict__ p, int n) {
    int i = blockIdx.x * blockDim.x + threadIdx.x;
    if (i < n) p[i] = 0.0f;
}

// One wave32 per edge; each lane moves a float4 (128 floats / edge).
__global__ void __launch_bounds__(256)
scatter_kernel(const float* __restrict__ x,
               const long long* __restrict__ ei,
               float* __restrict__ agg) {
    int gid  = blockIdx.x * 256 + threadIdx.x;
    int edge = gid >> 5;
    int lane = gid & 31;
    long long s = ei[edge];
    long long d = ei[EDGES + edge];
    float4 v = ((const float4*)(x + (size_t)s * DIM))[lane];
    float* dp = agg + (size_t)d * DIM + lane * 4;
    atomicAdd(dp + 0, v.x);
    atomicAdd(dp + 1, v.y);
    atomicAdd(dp + 2, v.z);
    atomicAdd(dp + 3, v.w);
}

// Fused: h = (1+eps)*x + agg (staged to LDS), then h @ W + b (+ReLU) via
// v_wmma_f32_16x16x4_f32, chained over K=128 in 32 steps. One block = 16 rows,
// 8 waves = 8 column tiles of 16. In-place safe (rows block-private).
__global__ void __launch_bounds__(256)
gin_gemm_kernel(const float* __restrict__ xin,
                const float* __restrict__ agg,
                const float* __restrict__ W,     // [128,128] row-major
                const float* __restrict__ bias,  // [128]
                const float* __restrict__ eps_p, // scalar
                float* __restrict__ xout,
                int do_relu) {
    __shared__ float hs[16 * 132];  // stride 132: bank-conflict-free A loads
    const int tid = threadIdx.x;
    const int rowBase = blockIdx.x * 16;
    const float ep = 1.0f + *eps_p;

    // Stage h tile: 16 rows x 128 cols, 8 floats per thread, coalesced.
    {
        int r = tid >> 4;
        int c = (tid & 15) * 8;
        const float4* xr = (const float4*)(xin + (size_t)(rowBase + r) * DIM + c);
        const float4* ar = (const float4*)(agg + (size_t)(rowBase + r) * DIM + c);
        float4 x0 = xr[0], x1 = xr[1];
        float4 a0 = ar[0], a1 = ar[1];
        float* hp = &hs[r * 132 + c];
        hp[0] = ep * x0.x + a0.x;  hp[1] = ep * x0.y + a0.y;
        hp[2] = ep * x0.z + a0.z;  hp[3] = ep * x0.w + a0.w;
        hp[4] = ep * x1.x + a1.x;  hp[5] = ep * x1.y + a1.y;
        hp[6] = ep * x1.z + a1.z;  hp[7] = ep * x1.w + a1.w;
    }
    __syncthreads();

    const int wave = tid >> 5;        // column tile 0..7
    const int lane = tid & 31;
    const int m    = lane & 15;       // row within tile / N within col tile
    const int half = lane >> 4;       // K-slice selector per ISA layout
    const int koff = half * 2;
    const int n    = wave * 16 + m;   // output column

    v8f acc = {};
    #pragma unroll
    for (int k = 0; k < DIM; k += 4) {
        // A (16x4 f32): lanes 0-15 hold K=k..k+1, lanes 16-31 hold K=k+2..k+3
        v2f a = *(const v2f*)&hs[m * 132 + k + koff];
        // B (4x16 f32): row-striped across lanes, same half-wave K split
        v2f bf;
        bf.x = W[(k + koff    ) * DIM + n];
        bf.y = W[(k + koff + 1) * DIM + n];
        acc = __builtin_amdgcn_wmma_f32_16x16x4_f32(false, a, false, bf,
                                                    (short)0, acc, false, false);
    }

    const float bv = bias[n];
    #pragma unroll
    for (int r = 0; r < 8; ++r) {
        // D layout: VGPR r -> row r (lanes 0-15) / row r+8 (lanes 16-31)
        float v = acc[r] + bv;
        if (do_relu) v = fmaxf(v, 0.0f);
        xout[(size_t)(rowBase + r + half * 8) * DIM + n] = v;
    }
}

// One wave32 per node: pool rows into out[batch[node]].
__global__ void __launch_bounds__(256)
pool_kernel(const float* __restrict__ x,
            const long long* __restrict__ batch,
            float* __restrict__ out) {
    int gid  = blockIdx.x * 256 + threadIdx.x;
    int node = gid >> 5;
    int lane = gid & 31;
    if (node >= NODES) return;
    long long g = batch[node];
    float4 v = ((const float4*)(x + (size_t)node * DIM))[lane];
    float* o = out + (size_t)g * DIM + lane * 4;
    atomicAdd(o + 0, v.x);
    atomicAdd(o + 1, v.y);
    atomicAdd(o + 2, v.z);
    atomicAdd(o + 3, v.w);
}

extern "C" void kernel_launch(void* const* d_in, const int* in_sizes, int n_in,
                              void* d_out, int out_size, void* d_ws, size_t ws_size,
                              hipStream_t stream) {
    const float*     x     = (const float*)d_in[0];
    const long long* ei    = (const long long*)d_in[1];
    const long long* batch = (const long long*)d_in[2];
    const float*     W     = (const float*)d_in[3];
    const float*     b     = (const float*)d_in[4];
    const float*     eps   = (const float*)d_in[5];
    float*           out   = (float*)d_out;

    float* agg  = (float*)d_ws;                  // NODES*DIM f32 (25.6 MB)
    float* xcur = agg + (size_t)NODES * DIM;     // NODES*DIM f32 (25.6 MB)

    const int NF = NODES * DIM;

    zero_kernel<<<(GRAPHS * DIM + 255) / 256, 256, 0, stream>>>(out, GRAPHS * DIM);

    for (int layer = 0; layer < LAYERS; ++layer) {
        const float* xin = (layer == 0) ? x : xcur;
        zero_kernel<<<(NF + 255) / 256, 256, 0, stream>>>(agg, NF);
        scatter_kernel<<<(EDGES * 32) / 256, 256, 0, stream>>>(xin, ei, agg);
        gin_gemm_kernel<<<NODES / 16, 256, 0, stream>>>(
            xin, agg,
            W + (size_t)layer * DIM * DIM,
            b + (size_t)layer * DIM,
            eps + layer,
            xcur,
            (layer < LAYERS - 1) ? 1 : 0);
    }

    pool_kernel<<<(NODES * 32 + 255) / 256, 256, 0, stream>>>(xcur, batch, out);
}